// SelectBlock_80994493268152
// MI455X (gfx1250) — compile-verified
//
#include <hip/hip_runtime.h>
#include <hip/hip_bf16.h>
#include <math.h>

// Problem constants
#define BB 16
#define SS 4096
#define DD 2048
#define HH 16
#define HD 128
#define DFF 8192
#define NSPLIT 17        // 16 splits of 256 cached tokens + 1 for the new token

typedef __attribute__((ext_vector_type(2))) float v2f;
typedef __attribute__((ext_vector_type(8))) float v8f;

// Workspace layout (float offsets), all multiples of 32
#define OFF_QKV      0          // 16*6144
#define OFF_MLPLOG   98304      // 16*8192
#define OFF_HEADLOG  229376     // 256
#define OFF_HEADMASK 229632     // 256
#define OFF_THRESH   229888     // 16 (as unsigned)
#define OFF_PART     229920     // 16*16*17*130 = 565760
#define OFF_CTX      795680     // 16*2048
#define OFF_H        828448     // 16*2048
#define OFF_A        861216     // 16*8192

__device__ __forceinline__ unsigned ord_f32(float f) {
    unsigned u = __float_as_uint(f);
    // monotone map: float order -> unsigned order
    return u ^ ((((int)u) >> 31) | 0x80000000u);
}

// ---------------------------------------------------------------------------
// Generic M=16 fp32 GEMM via V_WMMA_F32_16X16X4_F32.
// Out[16 x N] = X[16 x K] @ W[K x N]  (+ epilogue per mode)
// mode 0: plain store
// mode 1: + resid[16 x N]
// mode 2: relu(acc + bias[col]) masked by (ord(logits[row*N+col]) >= thresh[row])
// mode 3: + bias[col]
// One wave per 16-column output tile; 8 waves per block.
// ---------------------------------------------------------------------------
__global__ void gemm_m16_wmma(const float* __restrict__ X, const float* __restrict__ W,
                              float* __restrict__ Out, int K, int N, int mode,
                              const float* __restrict__ bias,
                              const float* __restrict__ resid,
                              const float* __restrict__ logits,
                              const unsigned* __restrict__ thresh) {
    int wave = threadIdx.x >> 5;
    int tile = blockIdx.x * (blockDim.x >> 5) + wave;
    int n0 = tile * 16;
    if (n0 >= N) return;                 // wave-uniform: EXEC stays all-ones
    int lane = threadIdx.x & 31;
    int half = lane >> 4;                // 0: lanes 0-15, 1: lanes 16-31
    int l    = lane & 15;

    // A fragment (16x4 f32): lane = M row; VGPR0 = K{0,2}, VGPR1 = K{1,3}
    // B fragment (4x16 f32): lane = N col; VGPR0 = K{0,2}, VGPR1 = K{1,3}
    const float* xrow = X + (size_t)l * K;
    const float* wcol = W + n0 + l;

    v8f acc0 = {};
    v8f acc1 = {};
    for (int k0 = 0; k0 < K; k0 += 8) {
        int ka = k0 + 2 * half;
        v2f a0 = *(const v2f*)(xrow + ka);          // K = ka, ka+1
        v2f b0;
        b0.x = wcol[(size_t)ka * N];
        b0.y = wcol[(size_t)(ka + 1) * N];
        acc0 = __builtin_amdgcn_wmma_f32_16x16x4_f32(false, a0, false, b0,
                                                     (short)0, acc0, false, false);
        int kb = k0 + 4 + 2 * half;
        v2f a1 = *(const v2f*)(xrow + kb);
        v2f b1f;
        b1f.x = wcol[(size_t)kb * N];
        b1f.y = wcol[(size_t)(kb + 1) * N];
        acc1 = __builtin_amdgcn_wmma_f32_16x16x4_f32(false, a1, false, b1f,
                                                     (short)0, acc1, false, false);
    }

    // C/D layout: VGPR r holds rows (r, r+8); lanes 0-15 / 16-31 are N cols
    int col = n0 + l;
    #pragma unroll
    for (int r = 0; r < 8; ++r) {
        int row = r + half * 8;
        float v = acc0[r] + acc1[r];
        if (mode == 1) {
            v += resid[(size_t)row * N + col];
        } else if (mode == 2) {
            v = fmaxf(v + bias[col], 0.0f);
            if (ord_f32(logits[(size_t)row * N + col]) < thresh[row]) v = 0.0f;
        } else if (mode == 3) {
            v += bias[col];
        }
        Out[(size_t)row * N + col] = v;
    }
}

// ---------------------------------------------------------------------------
// Top-k masks: block 0 -> head mask (exact rank with index tie-break, like
// jax.lax.top_k); blocks 1..16 -> per-row radix select threshold for MLP.
// ---------------------------------------------------------------------------
__global__ void masks_kernel(const float* __restrict__ head_logits,
                             const float* __restrict__ mlp_logits,
                             const int* __restrict__ attn_topk,
                             const int* __restrict__ mlp_topk,
                             float* __restrict__ head_mask,
                             unsigned* __restrict__ thresh) {
    if (blockIdx.x == 0) {
        int t = threadIdx.x;            // 256 = 16 rows x 16 heads
        int b = t >> 4, i = t & 15;
        float v = head_logits[b * 16 + i];
        int rank = 0;
        for (int j = 0; j < 16; ++j) {
            float vj = head_logits[b * 16 + j];
            if (vj > v || (vj == v && j < i)) rank++;
        }
        head_mask[t] = (rank < attn_topk[0]) ? 1.0f : 0.0f;
        return;
    }
    int r = blockIdx.x - 1;
    const float* row = mlp_logits + (size_t)r * DFF;
    __shared__ unsigned cnt;
    int tid = threadIdx.x;
    unsigned prefix = 0;
    int kk = mlp_topk[0];
    for (int bit = 31; bit >= 0; --bit) {
        __syncthreads();
        if (tid == 0) cnt = 0;
        __syncthreads();
        unsigned hm = (bit == 31) ? 0u : (0xFFFFFFFFu << (bit + 1));
        unsigned local = 0;
        for (int n = tid; n < DFF; n += 256) {
            unsigned u = ord_f32(row[n]);
            if ((u & hm) == prefix && (u & (1u << bit))) local++;
        }
        atomicAdd(&cnt, local);
        __syncthreads();
        unsigned c = cnt;
        if ((int)c >= kk) prefix |= (1u << bit);
        else              kk -= (int)c;
    }
    if (tid == 0) thresh[r] = prefix;   // keep logits with ord >= prefix
}

// ---------------------------------------------------------------------------
// Flash-style split attention. One block per (b, h, split).
// split < 16: 256 cached positions; split == 16: the new token from qkv.
// Emits per-split (m, l, 128-dim unnormalized ctx).
// ---------------------------------------------------------------------------
__global__ void attn_partial_kernel(const float* __restrict__ qkv,
                                    const float* __restrict__ kc,
                                    const float* __restrict__ vc,
                                    float* __restrict__ part) {
    int split = blockIdx.x % NSPLIT;
    int bh    = blockIdx.x / NSPLIT;
    int b = bh >> 4, h = bh & 15;
    int tid = threadIdx.x;

    __shared__ float q_sh[HD];
    __shared__ float sc[256];
    __shared__ float red[256];

    if (tid < HD) q_sh[tid] = qkv[(size_t)b * 6144 + h * HD + tid] * 0.08838834764831843f;
    __syncthreads();

    float* out = part + (size_t)(bh * NSPLIT + split) * 130;

    if (split < 16) {
        int wv = tid >> 5, lane = tid & 31;
        // wave wv handles 32 consecutive positions; 32 lanes cooperate per row
        for (int i = 0; i < 32; ++i) {
            int idx = wv * 32 + i;
            int pos = split * 256 + idx;
            const float* kr = kc + ((((size_t)b * SS + pos) * HH + h) << 7);
            float4 k4 = ((const float4*)kr)[lane];   // 512B coalesced per wave
            float d = q_sh[lane * 4 + 0] * k4.x + q_sh[lane * 4 + 1] * k4.y
                    + q_sh[lane * 4 + 2] * k4.z + q_sh[lane * 4 + 3] * k4.w;
            for (int off = 16; off; off >>= 1) d += __shfl_xor(d, off, 32);
            if (lane == 0) sc[idx] = d;
        }
        __syncthreads();
        // block max
        red[tid] = sc[tid]; __syncthreads();
        for (int s = 128; s > 0; s >>= 1) {
            if (tid < s) red[tid] = fmaxf(red[tid], red[tid + s]);
            __syncthreads();
        }
        float m = red[0]; __syncthreads();
        float e = __expf(sc[tid] - m);
        sc[tid] = e;                      // scores -> weights
        red[tid] = e; __syncthreads();
        for (int s = 128; s > 0; s >>= 1) {
            if (tid < s) red[tid] += red[tid + s];
            __syncthreads();
        }
        float lsum = red[0];
        __syncthreads();
        if (tid < HD) {
            float acc = 0.0f;
            for (int p = 0; p < 256; ++p) {
                const float* vr = vc + ((((size_t)b * SS + split * 256 + p) * HH + h) << 7);
                acc += sc[p] * vr[tid];   // 512B coalesced per iteration
            }
            out[2 + tid] = acc;
        }
        if (tid == 0) { out[0] = m; out[1] = lsum; }
    } else {
        // new token: k_new at qkv comp 1, v_new at comp 2
        if (tid < 32) {
            int lane = tid;
            const float* kn = qkv + (size_t)b * 6144 + 2048 + h * HD;
            float4 k4 = ((const float4*)kn)[lane];
            float d = q_sh[lane * 4 + 0] * k4.x + q_sh[lane * 4 + 1] * k4.y
                    + q_sh[lane * 4 + 2] * k4.z + q_sh[lane * 4 + 3] * k4.w;
            for (int off = 16; off; off >>= 1) d += __shfl_xor(d, off, 32);
            if (lane == 0) sc[0] = d;
        }
        __syncthreads();
        if (tid == 0) { out[0] = sc[0]; out[1] = 1.0f; }
        if (tid < HD) out[2 + tid] = qkv[(size_t)b * 6144 + 4096 + h * HD + tid];
    }
}

// Combine splits, normalize, apply head mask -> ctx (16 x 2048)
__global__ void attn_combine_kernel(const float* __restrict__ part,
                                    const float* __restrict__ head_mask,
                                    float* __restrict__ ctx) {
    int bh = blockIdx.x;
    int t = threadIdx.x;                  // 128
    const float* p0 = part + (size_t)bh * NSPLIT * 130;
    float M = -3.4e38f;
    for (int i = 0; i < NSPLIT; ++i) M = fmaxf(M, p0[i * 130]);
    float L = 0.0f, acc = 0.0f;
    for (int i = 0; i < NSPLIT; ++i) {
        float w = __expf(p0[i * 130] - M);
        L   += p0[i * 130 + 1] * w;
        acc += p0[i * 130 + 2 + t] * w;
    }
    ctx[(size_t)bh * HD + t] = acc / L * head_mask[bh];
}

// LayerNorm over D=2048, one block per batch row
__global__ void layernorm_kernel(const float* __restrict__ xin,
                                 const float* __restrict__ w,
                                 const float* __restrict__ bias,
                                 float* __restrict__ out) {
    int b = blockIdx.x, tid = threadIdx.x;
    __shared__ float red[256];
    const float* x = xin + (size_t)b * DD;
    float s = 0.0f;
    for (int i = tid; i < DD; i += 256) s += x[i];
    red[tid] = s; __syncthreads();
    for (int st = 128; st > 0; st >>= 1) {
        if (tid < st) red[tid] += red[tid + st];
        __syncthreads();
    }
    float mu = red[0] / (float)DD; __syncthreads();
    float s2 = 0.0f;
    for (int i = tid; i < DD; i += 256) { float d = x[i] - mu; s2 += d * d; }
    red[tid] = s2; __syncthreads();
    for (int st = 128; st > 0; st >>= 1) {
        if (tid < st) red[tid] += red[tid + st];
        __syncthreads();
    }
    float rstd = rsqrtf(red[0] / (float)DD + 1e-5f);
    for (int i = tid; i < DD; i += 256)
        out[(size_t)b * DD + i] = (x[i] - mu) * rstd * w[i] + bias[i];
}

extern "C" void kernel_launch(void* const* d_in, const int* in_sizes, int n_in,
                              void* d_out, int out_size, void* d_ws, size_t ws_size,
                              hipStream_t stream) {
    const float* x       = (const float*)d_in[0];   // (16, 1, 2048)
    const float* resid   = (const float*)d_in[1];   // (16, 1, 2048)
    const float* k_cache = (const float*)d_in[2];
    const float* v_cache = (const float*)d_in[3];
    const float* Wqkv    = (const float*)d_in[4];
    const float* Wo      = (const float*)d_in[5];
    const float* Wmha    = (const float*)d_in[6];
    const float* Wmlp    = (const float*)d_in[7];
    const float* W1      = (const float*)d_in[8];
    const float* b1      = (const float*)d_in[9];
    const float* W2      = (const float*)d_in[10];
    const float* b2      = (const float*)d_in[11];
    const float* norm_w  = (const float*)d_in[12];
    const float* norm_b  = (const float*)d_in[13];
    const int* mlp_topk  = (const int*)d_in[14];
    const int* attn_topk = (const int*)d_in[15];

    float* ws = (float*)d_ws;
    float*    qkv       = ws + OFF_QKV;
    float*    mlp_log   = ws + OFF_MLPLOG;
    float*    head_log  = ws + OFF_HEADLOG;
    float*    head_mask = ws + OFF_HEADMASK;
    unsigned* thresh    = (unsigned*)(ws + OFF_THRESH);
    float*    part      = ws + OFF_PART;
    float*    ctx       = ws + OFF_CTX;
    float*    hbuf      = ws + OFF_H;
    float*    abuf      = ws + OFF_A;

    float* out_mlp   = (float*)d_out;            // (16, 1, 2048)
    float* resid_out = out_mlp + BB * DD;        // (16, 1, 2048)

    // 1-3: routers + qkv  (x @ W)
    gemm_m16_wmma<<<1, 256, 0, stream>>>(x, Wmha, head_log, DD, 16, 0,
                                         nullptr, nullptr, nullptr, nullptr);
    gemm_m16_wmma<<<48, 256, 0, stream>>>(x, Wqkv, qkv, DD, 3 * DD, 0,
                                          nullptr, nullptr, nullptr, nullptr);
    gemm_m16_wmma<<<64, 256, 0, stream>>>(x, Wmlp, mlp_log, DD, DFF, 0,
                                          nullptr, nullptr, nullptr, nullptr);
    // 4: top-k masks
    masks_kernel<<<1 + BB, 256, 0, stream>>>(head_log, mlp_log, attn_topk,
                                             mlp_topk, head_mask, thresh);
    // 5-6: attention (dominant 1.07 GB of KV traffic)
    attn_partial_kernel<<<BB * HH * NSPLIT, 256, 0, stream>>>(qkv, k_cache, v_cache, part);
    attn_combine_kernel<<<BB * HH, 128, 0, stream>>>(part, head_mask, ctx);
    // 7: mixer = ctx @ Wo, residual add -> second half of d_out
    gemm_m16_wmma<<<16, 256, 0, stream>>>(ctx, Wo, resid_out, DD, DD, 1,
                                          nullptr, resid, nullptr, nullptr);
    // 8: layernorm
    layernorm_kernel<<<BB, 256, 0, stream>>>(resid_out, norm_w, norm_b, hbuf);
    // 9: a = relu(h @ W1 + b1) * neuron_mask
    gemm_m16_wmma<<<64, 256, 0, stream>>>(hbuf, W1, abuf, DD, DFF, 2,
                                          b1, nullptr, mlp_log, thresh);
    // 10: out = a @ W2 + b2 -> first half of d_out
    gemm_m16_wmma<<<16, 256, 0, stream>>>(abuf, W2, out_mlp, DFF, DD, 3,
                                          b2, nullptr, nullptr, nullptr);
}